// ZayaRouter_61830349193727
// MI455X (gfx1250) — compile-verified
//
#include <hip/hip_runtime.h>
#include <math.h>

#define S_TOK 32768
#define H_DIM 4096
#define D_DIM 1024
#define E_NUM 17
#define TOPK  4

typedef float v2f __attribute__((ext_vector_type(2)));
typedef float v8f __attribute__((ext_vector_type(8)));
typedef int   v4i __attribute__((ext_vector_type(4)));

#define EP_EDA     0   // x + bias[col] + prev[row,col]*eda[col]
#define EP_GELU    1   // gelu(x + bias[col])
#define EP_GELU_RS 2   // gelu(x*rowscale[row] + bias[col])   (RMSNorm row factor)

#ifndef __has_builtin
#define __has_builtin(x) 0
#endif
#if __has_builtin(__builtin_amdgcn_global_load_async_to_lds_b128)
#define HAVE_ASYNC_LDS 1
#else
#define HAVE_ASYNC_LDS 0
#endif

#define AS1 __attribute__((address_space(1)))
#define AS3 __attribute__((address_space(3)))

__device__ __forceinline__ void async_copy_b128(const float* g, float* l) {
#if HAVE_ASYNC_LDS
    __builtin_amdgcn_global_load_async_to_lds_b128(
        (AS1 v4i*)g, (AS3 v4i*)l, 0, 0);
#else
    *(float4*)l = *(const float4*)g;
#endif
}

__device__ __forceinline__ void wait_async_lds() {
#if HAVE_ASYNC_LDS
#if __has_builtin(__builtin_amdgcn_s_wait_asynccnt)
    __builtin_amdgcn_s_wait_asynccnt(0);
#else
    asm volatile("s_wait_asynccnt 0x0" ::: "memory");
#endif
#endif
}

// C[M,N] = A[M,K] @ W[N,K]^T (+ epilogue).  BM=64, BN=128, KC=32.
// Double-buffered LDS; async global->LDS staging when available.
template<int BM, int BN, int KC, int EPI>
__global__ __launch_bounds__(256) void wmma_gemm_kernel(
    const float* __restrict__ A, const float* __restrict__ W,
    const float* __restrict__ bias,
    const float* __restrict__ ex0,   // EP_EDA: prev (M*N); EP_GELU_RS: rowscale (M)
    const float* __restrict__ ex1,   // EP_EDA: eda_scale (N)
    float* __restrict__ C, int M, int N, int K)
{
    __shared__ __align__(16) float As[2][BM][KC + 4];
    __shared__ __align__(16) float Bs[2][BN][KC + 4];

    const int t    = threadIdx.x;
    const int lane = t & 31;
    const int wave = t >> 5;
    const int wm   = wave & 1;     // 2 waves along M
    const int wn   = wave >> 1;    // 4 waves along N
    const int row0 = blockIdx.y * BM;
    const int col0 = blockIdx.x * BN;

    const int lm    = lane & 15;
    const int khalf = (lane >> 4) << 1;   // lanes16-31 hold K+2/K+3
    const int chalf = (lane >> 4) << 3;   // C/D: lanes16-31 hold rows M+8..15

    v8f acc[2][2] = {};

    auto stage = [&](int buf, int k0) {
        #pragma unroll
        for (int i = 0; i < (BM * KC / 4) / 256; ++i) {
            int idx = t + i * 256;
            int r   = idx >> 3;            // KC/4 = 8 float4 per row
            int c4  = (idx & 7) << 2;
            async_copy_b128(A + (long)(row0 + r) * K + k0 + c4, &As[buf][r][c4]);
        }
        #pragma unroll
        for (int i = 0; i < (BN * KC / 4) / 256; ++i) {
            int idx = t + i * 256;
            int r   = idx >> 3;
            int c4  = (idx & 7) << 2;
            async_copy_b128(W + (long)(col0 + r) * K + k0 + c4, &Bs[buf][r][c4]);
        }
    };

    stage(0, 0);

    for (int k0 = 0, it = 0; k0 < K; k0 += KC, ++it) {
        const int cur = it & 1;
        wait_async_lds();        // own async stores to LDS complete
        __syncthreads();         // whole tile visible; prior reads of buf done
        if (k0 + KC < K) stage(cur ^ 1, k0 + KC);

        const float (*Ac)[KC + 4] = As[cur];
        const float (*Bc)[KC + 4] = Bs[cur];
        #pragma unroll
        for (int k4 = 0; k4 < KC; k4 += 4) {
            const int kk = k4 + khalf;
            v2f a0, a1, b0, b1;
            a0.x = Ac[wm * 32 +  0 + lm][kk];  a0.y = Ac[wm * 32 +  0 + lm][kk + 1];
            a1.x = Ac[wm * 32 + 16 + lm][kk];  a1.y = Ac[wm * 32 + 16 + lm][kk + 1];
            b0.x = Bc[wn * 32 +  0 + lm][kk];  b0.y = Bc[wn * 32 +  0 + lm][kk + 1];
            b1.x = Bc[wn * 32 + 16 + lm][kk];  b1.y = Bc[wn * 32 + 16 + lm][kk + 1];
            acc[0][0] = __builtin_amdgcn_wmma_f32_16x16x4_f32(false, a0, false, b0, (short)0, acc[0][0], false, false);
            acc[0][1] = __builtin_amdgcn_wmma_f32_16x16x4_f32(false, a0, false, b1, (short)0, acc[0][1], false, false);
            acc[1][0] = __builtin_amdgcn_wmma_f32_16x16x4_f32(false, a1, false, b0, (short)0, acc[1][0], false, false);
            acc[1][1] = __builtin_amdgcn_wmma_f32_16x16x4_f32(false, a1, false, b1, (short)0, acc[1][1], false, false);
        }
        __syncthreads();
    }

    // ---- epilogue ----
    #pragma unroll
    for (int mi = 0; mi < 2; ++mi) {
        #pragma unroll
        for (int ni = 0; ni < 2; ++ni) {
            const int colg = col0 + wn * 32 + ni * 16 + lm;
            const float bcol = bias[colg];
            float edac = 0.0f;
            if constexpr (EPI == EP_EDA) edac = ex1[colg];
            #pragma unroll
            for (int v = 0; v < 8; ++v) {
                const int rowg = row0 + wm * 32 + mi * 16 + chalf + v;
                float x = acc[mi][ni][v];
                if constexpr (EPI == EP_EDA) {
                    x += bcol + ex0[(long)rowg * N + colg] * edac;
                } else {
                    if constexpr (EPI == EP_GELU_RS) x *= ex0[rowg];
                    x += bcol;
                    x = 0.5f * x * (1.0f + erff(x * 0.70710678118654752f));
                }
                C[(long)rowg * N + colg] = x;
            }
        }
    }
}

// W1s[e,d] = W1[e,d] * rms_w[d]   (folds RMSNorm channel weight into W1)
__global__ __launch_bounds__(256) void scale_w1_kernel(
    const float* __restrict__ W1, const float* __restrict__ rmsw,
    float* __restrict__ W1s)
{
    const long i4 = (long)blockIdx.x * 256 + threadIdx.x;   // float4 index
    float4 w = ((const float4*)W1)[i4];
    const int d = (int)((i4 << 2) & (D_DIM - 1));
    w.x *= rmsw[d];     w.y *= rmsw[d + 1];
    w.z *= rmsw[d + 2]; w.w *= rmsw[d + 3];
    ((float4*)W1s)[i4] = w;
}

// scale[row] = rsqrt(mean(hs[row,:]^2) + 1e-5), D = 1024
__global__ __launch_bounds__(256) void rms_scale_kernel(
    const float* __restrict__ hs, float* __restrict__ scale)
{
    __shared__ float red[8];
    const int row = blockIdx.x;
    const float4 v = ((const float4*)(hs + (long)row * D_DIM))[threadIdx.x];
    float s = v.x * v.x + v.y * v.y + v.z * v.z + v.w * v.w;
    #pragma unroll
    for (int off = 16; off > 0; off >>= 1) s += __shfl_xor(s, off, 32);
    if ((threadIdx.x & 31) == 0) red[threadIdx.x >> 5] = s;
    __syncthreads();
    if (threadIdx.x == 0) {
        float tot = 0.f;
        #pragma unroll
        for (int i = 0; i < 8; ++i) tot += red[i];
        scale[row] = rsqrtf(tot * (1.0f / D_DIM) + 1e-5f);
    }
}

// logits[tok,e] = h2[tok,:] . W3[e,:]   (one wave per token, E=17)
__global__ __launch_bounds__(256) void logits_kernel(
    const float* __restrict__ h2, const float* __restrict__ W3,
    float* __restrict__ logits)
{
    const int lane = threadIdx.x & 31;
    const int tok  = blockIdx.x * 8 + (threadIdx.x >> 5);
    const float* hrow = h2 + (long)tok * D_DIM;
    float hreg[32];
    #pragma unroll
    for (int i = 0; i < 32; ++i) hreg[i] = hrow[lane + i * 32];
    #pragma unroll 1
    for (int e = 0; e < E_NUM; ++e) {
        const float* wrow = W3 + (long)e * D_DIM;
        float s = 0.f;
        #pragma unroll
        for (int i = 0; i < 32; ++i) s = fmaf(hreg[i], wrow[lane + i * 32], s);
        #pragma unroll
        for (int off = 16; off > 0; off >>= 1) s += __shfl_xor(s, off, 32);
        if (lane == 0) logits[(long)tok * E_NUM + e] = s;
    }
}

// fp32 softmax -> biased top-4 (lowest-index ties) -> MoD mask -> gather
__global__ __launch_bounds__(256) void router_topk_kernel(
    const float* __restrict__ logits, const float* __restrict__ bb,
    float* __restrict__ out_prob, long long* __restrict__ out_choice)
{
    const int tok = blockIdx.x * 256 + threadIdx.x;
    float l[E_NUM];
    float m = -1e30f;
    #pragma unroll
    for (int e = 0; e < E_NUM; ++e) {
        l[e] = logits[(long)tok * E_NUM + e];
        m = fmaxf(m, l[e]);
    }
    float sum = 0.f;
    #pragma unroll
    for (int e = 0; e < E_NUM; ++e) { l[e] = expf(l[e] - m); sum += l[e]; }
    const float inv = 1.0f / sum;
    float prob[E_NUM], biased[E_NUM];
    #pragma unroll
    for (int e = 0; e < E_NUM; ++e) {
        prob[e]   = l[e] * inv;
        biased[e] = prob[e] + bb[e];
    }
    int choice[TOPK];
    #pragma unroll
    for (int k = 0; k < TOPK; ++k) {
        int arg = 0; float best = -1e30f;
        #pragma unroll
        for (int e = 0; e < E_NUM; ++e)
            if (biased[e] > best) { best = biased[e]; arg = e; }
        choice[k] = arg;
        biased[arg] = -1e30f;
    }
    bool skip = false;
    #pragma unroll
    for (int k = 0; k < TOPK; ++k) {
        if (choice[k] == E_NUM - 1) skip = true;
        if (skip) choice[k] = E_NUM - 1;
        out_prob[(long)tok * TOPK + k]   = prob[choice[k]];
        out_choice[(long)tok * TOPK + k] = (long long)choice[k];
    }
}

extern "C" void kernel_launch(void* const* d_in, const int* in_sizes, int n_in,
                              void* d_out, int out_size, void* d_ws, size_t ws_size,
                              hipStream_t stream) {
    const int S = S_TOK, H = H_DIM, D = D_DIM;
    const float* X    = (const float*)d_in[0];
    const float* prev = (const float*)d_in[1];
    const float* Wd   = (const float*)d_in[2];
    const float* bd   = (const float*)d_in[3];
    const float* rmsw = (const float*)d_in[4];
    const float* eda  = (const float*)d_in[5];
    const float* W1   = (const float*)d_in[6];
    const float* b1   = (const float*)d_in[7];
    const float* W2   = (const float*)d_in[8];
    const float* b2   = (const float*)d_in[9];
    const float* W3   = (const float*)d_in[10];
    const float* bb   = (const float*)d_in[11];

    // d_out: route_prob (S*4 f32) | expert_choice (S*4 i64) | hs_next (S*D f32)
    float*      out_prob   = (float*)d_out;
    long long*  out_choice = (long long*)((char*)d_out + (size_t)S * TOPK * sizeof(float));
    float*      out_hs     = (float*)((char*)d_out +
                              (size_t)S * TOPK * (sizeof(float) + sizeof(long long)));

    // workspace: rowscale (S) | W1*rms_w (D*D) | h1 (S*D) | h2 (S*D) | logits (S*E)
    float* ws_scale  = (float*)d_ws;
    float* ws_w1s    = ws_scale + S;
    float* ws_h1     = ws_w1s + (size_t)D * D;
    float* ws_h2     = ws_h1 + (size_t)S * D;
    float* ws_logits = ws_h2 + (size_t)S * D;

    const dim3 blk(256);
    const dim3 gemm_grid(D / 128, S / 64);

    // 0) fold rms_w into W1 (independent of the big GEMM)
    scale_w1_kernel<<<dim3((D * D / 4) / 256), blk, 0, stream>>>(W1, rmsw, ws_w1s);
    // 1) hs = X @ Wd^T + bd + prev*eda  -> out_hs (also the 3rd output)
    wmma_gemm_kernel<64, 128, 32, EP_EDA>
        <<<gemm_grid, blk, 0, stream>>>(X, Wd, bd, prev, eda, out_hs, S, D, H);
    // 2) per-row rsqrt(mean square + eps)
    rms_scale_kernel<<<dim3(S), blk, 0, stream>>>(out_hs, ws_scale);
    // 3) h1 = gelu(rs[row] * (hs @ (W1*rms_w)^T) + b1)  ==  gelu(RMSNorm(hs) @ W1^T + b1)
    wmma_gemm_kernel<64, 128, 32, EP_GELU_RS>
        <<<gemm_grid, blk, 0, stream>>>(out_hs, ws_w1s, b1, ws_scale, nullptr, ws_h1, S, D, D);
    // 4) h2 = gelu(h1 @ W2^T + b2)
    wmma_gemm_kernel<64, 128, 32, EP_GELU>
        <<<gemm_grid, blk, 0, stream>>>(ws_h1, W2, b2, nullptr, nullptr, ws_h2, S, D, D);
    // 5) logits = h2 @ W3^T  (E=17)
    logits_kernel<<<dim3(S / 8), blk, 0, stream>>>(ws_h2, W3, ws_logits);
    // 6) softmax + biased top-4 + MoD mask + gather
    router_topk_kernel<<<dim3(S / 256), blk, 0, stream>>>(ws_logits, bb, out_prob, out_choice);
}